// Rule_47450798686485
// MI455X (gfx1250) — compile-verified
//
#include <hip/hip_runtime.h>
#include <cstdint>

// Ising sweep: (B=16, C=1, H=2048, W=2048) fp32, radius-1 periodic stencil.
// HBM-bound (~822 MB/call -> ~35us @ 23.3 TB/s). TDM gather-mode tensor load
// stages a 10-row tile (8 computed rows + 2 halo, H-wrap folded into the row
// index list) into LDS in one DMA; waves then do coalesced float4 work.

typedef unsigned int u32x4 __attribute__((ext_vector_type(4)));
typedef int          i32x4 __attribute__((ext_vector_type(4)));
typedef int          i32x8 __attribute__((ext_vector_type(8)));

#define ISING_W   2048
#define ISING_H   2048
#define ISING_B   16
#define ROWS_PER_BLOCK 8                       // computed rows per block
#define TILE_ROWS (ROWS_PER_BLOCK + 2)         // + top/bottom halo
#define LDS_BYTES (TILE_ROWS * ISING_W * 4)    // 80 KB

__global__ __launch_bounds__(256) void ising_sweep_kernel(
    const float* __restrict__ x,
    const float* __restrict__ rnd,
    const float* __restrict__ drop,
    float* __restrict__ out)
{
    extern __shared__ float lds[];             // TILE_ROWS * 2048 floats

    const int blk = blockIdx.x;
    const int b   = blk >> 8;                  // 256 row-blocks per image
    const int h0  = (blk & 255) << 3;          // first computed row

    // ---- TDM descriptor (D#), gather mode, 16-bit row indices ----
    const unsigned long long base =
        (unsigned long long)(uintptr_t)x +
        (unsigned long long)b * (unsigned long long)(ISING_H * ISING_W) * 4ull;
    const unsigned int lds_off = (unsigned int)(uintptr_t)(&lds[0]);

    // Group 0: count=1, gather_mode=1 (bit31), 16-bit indices (bit30=0),
    //          lds_addr, global_addr[56:0], type=2 in [127:126].
    u32x4 g0;
    g0.x = 0x80000001u;
    g0.y = lds_off;
    g0.z = (unsigned int)(base & 0xFFFFFFFFull);
    g0.w = (unsigned int)((base >> 32) & 0x01FFFFFFull) | (2u << 30);

    // Group 1: workgroup_mask=0, data_size=2 (4B); tensor_dim0=W,
    // tensor_dim1=H, tile_dim0=W, tile_dim1=TILE_ROWS valid gather indices,
    // tensor_dim0_stride=W.
    i32x8 g1;
    g1[0] = 0x00020000;                         // data_size = 4 bytes
    g1[1] = (int)((unsigned)ISING_W << 16);     // tensor_dim0[15:0] @ [63:48]
    g1[2] = (int)((unsigned)ISING_H << 16);     // dim0[31:16]=0 | tensor_dim1[15:0]
    g1[3] = (int)((unsigned)ISING_W << 16);     // dim1[31:16]=0 | tile_dim0
    g1[4] = TILE_ROWS;                          // tile_dim1 = #gather indices
    g1[5] = ISING_W;                            // tensor_dim0_stride low 32
    g1[6] = 0;
    g1[7] = 0;

    // Groups 2/3: 16-bit row indices idx[r] = (h0 - 1 + r) mod H, r = 0..9.
    unsigned idx[TILE_ROWS];
    #pragma unroll
    for (int r = 0; r < TILE_ROWS; ++r)
        idx[r] = (unsigned)((h0 + (ISING_H - 1) + r) & (ISING_H - 1));

    i32x4 g2;
    g2[0] = (int)(idx[0] | (idx[1] << 16));
    g2[1] = (int)(idx[2] | (idx[3] << 16));
    g2[2] = (int)(idx[4] | (idx[5] << 16));
    g2[3] = (int)(idx[6] | (idx[7] << 16));
    i32x4 g3;
    g3[0] = (int)(idx[8] | (idx[9] << 16));
    g3[1] = 0; g3[2] = 0; g3[3] = 0;

    i32x8 g4 = {0, 0, 0, 0, 0, 0, 0, 0};       // extra operand (6-arg builtin)

    // One wave issues the DMA and waits on TENSORcnt; barrier releases the rest.
    if (threadIdx.x < 32) {
        __builtin_amdgcn_tensor_load_to_lds(g0, g1, g2, g3, g4, 0);
        __builtin_amdgcn_s_wait_tensorcnt(0);
    }
    __syncthreads();

    const float log2e_m2 = -2.0f * 1.44269504088896340736f; // fold 2x into exp2

    #pragma unroll 1
    for (int r = 0; r < ROWS_PER_BLOCK; ++r) {
        const float* __restrict__ up = lds + (size_t)r       * ISING_W;
        const float* __restrict__ ce = lds + (size_t)(r + 1) * ISING_W;
        const float* __restrict__ dn = lds + (size_t)(r + 2) * ISING_W;

        const int    h       = h0 + r;
        const size_t rowoff  = ((size_t)b * ISING_H + (size_t)h) * ISING_W;
        const size_t dropoff = (size_t)h * ISING_W;

        #pragma unroll
        for (int c = 0; c < 2; ++c) {
            const int w0 = ((int)threadIdx.x << 2) + c * (ISING_W / 2);

            const float4 rv = *(const float4*)(rnd  + rowoff  + w0);
            const float4 dv = *(const float4*)(drop + dropoff + w0);

            const float4 c4 = *(const float4*)(ce + w0);           // ds b128
            const float4 u4 = *(const float4*)(up + w0);           // ds b128
            const float4 d4 = *(const float4*)(dn + w0);           // ds b128
            const float  cl = ce[(w0 + ISING_W - 1) & (ISING_W - 1)];
            const float  cr = ce[(w0 + 4)           & (ISING_W - 1)];

            const float sA[4] = {c4.x, c4.y, c4.z, c4.w};
            const float lA[4] = {cl,   c4.x, c4.y, c4.z};
            const float rA[4] = {c4.y, c4.z, c4.w, cr  };
            const float uA[4] = {u4.x, u4.y, u4.z, u4.w};
            const float dA[4] = {d4.x, d4.y, d4.z, d4.w};
            const float* rvp = &rv.x;
            const float* dvp = &dv.x;

            float o[4];
            #pragma unroll
            for (int i = 0; i < 4; ++i) {
                const float s  = sA[i];
                const float m  = s * (lA[i] + rA[i] + uA[i] + dA[i]); // de = 2m
                // p = (de <= 0) ? 1 : exp(-de);  exp(-2m) = exp2(-2*log2e*m)
                const float p  = (m <= 0.0f) ? 1.0f
                                             : __builtin_amdgcn_exp2f(log2e_m2 * m);
                const bool flip = (rvp[i] < p) && (dvp[i] > 0.5f);
                o[i] = flip ? -s : s;
            }

            float4 ov;
            ov.x = o[0]; ov.y = o[1]; ov.z = o[2]; ov.w = o[3];
            *(float4*)(out + rowoff + w0) = ov;
        }
    }
}

extern "C" void kernel_launch(void* const* d_in, const int* in_sizes, int n_in,
                              void* d_out, int out_size, void* d_ws, size_t ws_size,
                              hipStream_t stream) {
    (void)in_sizes; (void)n_in; (void)d_ws; (void)ws_size; (void)out_size;
    const float* x    = (const float*)d_in[0];
    const float* rnd  = (const float*)d_in[1];
    const float* drop = (const float*)d_in[2];
    float* out        = (float*)d_out;

    const int blocks = ISING_B * (ISING_H / ROWS_PER_BLOCK);  // 4096
    ising_sweep_kernel<<<blocks, 256, LDS_BYTES, stream>>>(x, rnd, drop, out);
}